// MyGraphConv_29386166239373
// MI455X (gfx1250) — compile-verified
//
#include <hip/hip_runtime.h>
#include <cstdint>
#include <cstddef>

// Problem constants (from the reference).
#define N_NODES 50000
#define N_EDGES 800000
#define C       128
#define KEEP_SCALE 2.5f   // 1 / (1 - 0.6)

typedef __attribute__((ext_vector_type(2))) float v2f;
typedef __attribute__((ext_vector_type(8))) float v8f;

// ---------------------------------------------------------------------------
// h[i] = mask[i] ? x[i] * 2.5 : 0
// ---------------------------------------------------------------------------
__global__ void dropout_kernel(const float* __restrict__ x,
                               const unsigned char* __restrict__ mask,
                               float* __restrict__ h, int n) {
    int i = blockIdx.x * blockDim.x + threadIdx.x;
    if (i < n) h[i] = mask[i] ? x[i] * KEEP_SCALE : 0.0f;
}

// ---------------------------------------------------------------------------
// Zero the aggregation buffer (float4 stores).
// ---------------------------------------------------------------------------
__global__ void zero_kernel(float4* __restrict__ p, int n4) {
    int i = blockIdx.x * blockDim.x + threadIdx.x;
    if (i < n4) {
        float4 z; z.x = 0.f; z.y = 0.f; z.z = 0.f; z.w = 0.f;
        p[i] = z;
    }
}

// ---------------------------------------------------------------------------
// Edge scatter-add: one wave32 per edge, each lane moves 4 channels.
// agg[dst] += h[src]  via global_atomic_add_f32 (L2-resident: 25.6MB << 192MB).
// ---------------------------------------------------------------------------
__global__ __launch_bounds__(256)
void scatter_kernel(const float* __restrict__ h,
                    const long long* __restrict__ ei,
                    float* __restrict__ agg) {
    int t    = blockIdx.x * 256 + threadIdx.x;
    int e    = t >> 5;          // edge index (grid sized exactly: no guard)
    int lane = t & 31;
    int src = (int)ei[e];
    int dst = (int)ei[N_EDGES + e];
    const float4 v = *(const float4*)(h + (size_t)src * C + lane * 4);
    float* a = agg + (size_t)dst * C + lane * 4;
    atomicAdd(a + 0, v.x);
    atomicAdd(a + 1, v.y);
    atomicAdd(a + 2, v.z);
    atomicAdd(a + 3, v.w);
}

// ---------------------------------------------------------------------------
// Fused GraphConv GEMM tile kernel (one wave32 per 16x16 output tile):
//   out = agg @ Wrel^T + h @ Wroot^T + b   [+ ReLU] [+ next-layer dropout]
// Uses V_WMMA_F32_16X16X4_F32, K=128 in steps of 4, two accumulators to
// decouple the two matrix products and break the WMMA->WMMA RAW chain.
//
// A-fragment (16x4 f32, ISA 7.12.2): lanes 0-15 -> K=k..k+1, lanes 16-31 ->
// K=k+2..k+3, M = lane&15.  B = W^T, i.e. B[k][n] = W[n*128 + k], mirrored
// lane layout (N = lane&15, same K split).  C/D: VGPR r -> row r+8*(lane>=16),
// col lane&15.
// ---------------------------------------------------------------------------
template <bool RELU, bool DROP>
__global__ __launch_bounds__(256)
void gconv_wmma_kernel(const float* __restrict__ agg,
                       const float* __restrict__ h,
                       const float* __restrict__ Wrel,
                       const float* __restrict__ Wroot,
                       const float* __restrict__ bias,
                       const unsigned char* __restrict__ dropMask,
                       float* __restrict__ out) {
    const int wave = threadIdx.x >> 5;
    const int lane = threadIdx.x & 31;
    const int tile = blockIdx.x * 8 + wave;   // 25000 tiles total
    const int tm   = tile >> 3;               // node-tile  [0, 3125)
    const int tn   = tile & 7;                // chan-tile  [0, 8)
    const int r15  = lane & 15;
    const int kh   = (lane >> 4) << 1;        // 0 for lanes 0-15, 2 for 16-31

    const float* aggRow   = agg   + (size_t)(tm * 16 + r15) * C;
    const float* hRow     = h     + (size_t)(tm * 16 + r15) * C;
    const float* wrelRow  = Wrel  + (size_t)(tn * 16 + r15) * C;
    const float* wrootRow = Wroot + (size_t)(tn * 16 + r15) * C;

    v8f c0 = {};   // agg @ Wrel^T
    v8f c1 = {};   // h   @ Wroot^T
#pragma unroll
    for (int k = 0; k < C; k += 4) {
        v2f a0 = *(const v2f*)(aggRow   + k + kh);
        v2f b0 = *(const v2f*)(wrelRow  + k + kh);
        v2f a1 = *(const v2f*)(hRow     + k + kh);
        v2f b1 = *(const v2f*)(wrootRow + k + kh);
        c0 = __builtin_amdgcn_wmma_f32_16x16x4_f32(
                 false, a0, false, b0, (short)0, c0, false, false);
        c1 = __builtin_amdgcn_wmma_f32_16x16x4_f32(
                 false, a1, false, b1, (short)0, c1, false, false);
    }

    const float bv      = bias[tn * 16 + r15];
    const int   colOut  = tn * 16 + r15;
    const int   rowBase = tm * 16 + (lane >> 4) * 8;
#pragma unroll
    for (int r = 0; r < 8; ++r) {
        float v = c0[r] + c1[r] + bv;
        if (RELU) v = v > 0.0f ? v : 0.0f;
        size_t idx = (size_t)(rowBase + r) * C + colOut;
        if (DROP) out[idx] = dropMask[idx] ? v * KEEP_SCALE : 0.0f;
        else      out[idx] = v;
    }
}

// ---------------------------------------------------------------------------
// Host-side launch.  d_in order (setup_inputs dict):
//  0 x[N,128] f32 | 1 edge_index[2,E] i64 | 2..10 weights/biases f32
//  11 drop0[N,128] bool | 12 drop1 | 13 drop2
// ---------------------------------------------------------------------------
extern "C" void kernel_launch(void* const* d_in, const int* in_sizes, int n_in,
                              void* d_out, int out_size, void* d_ws, size_t ws_size,
                              hipStream_t stream) {
    const float*          x     = (const float*)d_in[0];
    const long long*      ei    = (const long long*)d_in[1];
    const float*          Wrel0 = (const float*)d_in[2];
    const float*          Wroot0= (const float*)d_in[3];
    const float*          b0    = (const float*)d_in[4];
    const float*          Wrel1 = (const float*)d_in[5];
    const float*          Wroot1= (const float*)d_in[6];
    const float*          b1    = (const float*)d_in[7];
    const float*          Wrel2 = (const float*)d_in[8];
    const float*          Wroot2= (const float*)d_in[9];
    const float*          b2    = (const float*)d_in[10];
    const unsigned char*  drop0 = (const unsigned char*)d_in[11];
    const unsigned char*  drop1 = (const unsigned char*)d_in[12];
    const unsigned char*  drop2 = (const unsigned char*)d_in[13];
    float*                out   = (float*)d_out;

    // Workspace layout: h_a | h_b | agg  (3 * 25.6 MB)
    float* h_a = (float*)d_ws;
    float* h_b = h_a + (size_t)N_NODES * C;
    float* agg = h_b + (size_t)N_NODES * C;

    const int NC = N_NODES * C;                 // 6,400,000
    const int dropBlocks    = (NC + 255) / 256;
    const int zeroBlocks    = (NC / 4 + 255) / 256;
    const int scatterBlocks = (N_EDGES * 32) / 256;   // 100,000 exact
    const int gemmBlocks    = (N_NODES / 16) * 8 / 8; // 3125 blocks * 8 waves

    // h0 = dropout(x)
    dropout_kernel<<<dropBlocks, 256, 0, stream>>>(x, drop0, h_a, NC);

    // ---- Layer 0: relu(conv(h0)) -> dropout(drop1) fused -> h_b
    zero_kernel<<<zeroBlocks, 256, 0, stream>>>((float4*)agg, NC / 4);
    scatter_kernel<<<scatterBlocks, 256, 0, stream>>>(h_a, ei, agg);
    gconv_wmma_kernel<true, true><<<gemmBlocks, 256, 0, stream>>>(
        agg, h_a, Wrel0, Wroot0, b0, drop1, h_b);

    // ---- Layer 1: relu(conv(h1)) -> dropout(drop2) fused -> h_a
    zero_kernel<<<zeroBlocks, 256, 0, stream>>>((float4*)agg, NC / 4);
    scatter_kernel<<<scatterBlocks, 256, 0, stream>>>(h_b, ei, agg);
    gconv_wmma_kernel<true, true><<<gemmBlocks, 256, 0, stream>>>(
        agg, h_b, Wrel1, Wroot1, b1, drop2, h_a);

    // ---- Layer 2: conv(h2) -> d_out (no relu, no dropout)
    zero_kernel<<<zeroBlocks, 256, 0, stream>>>((float4*)agg, NC / 4);
    scatter_kernel<<<scatterBlocks, 256, 0, stream>>>(h_a, ei, agg);
    gconv_wmma_kernel<false, false><<<gemmBlocks, 256, 0, stream>>>(
        agg, h_a, Wrel2, Wroot2, b2, nullptr, out);
}